// Attention_56916906606885
// MI455X (gfx1250) — compile-verified
//
#include <hip/hip_runtime.h>
#include <stdint.h>
#include <math.h>

#define B_DIM 32
#define C_DIM 512
#define HW    3136           // 56*56
#define K_SEL 256
#define NF4   784            // HW / 4 float4s per channel
#define NF4_MAIN 768         // staged via async copy (3 x 256 lanes)
#define NF4_TAIL 16          // remainder handled in registers

typedef float v4f __attribute__((ext_vector_type(4)));

__device__ __forceinline__ v4f vmax4(v4f a, v4f b) {
    v4f r;
    r.x = fmaxf(a.x, b.x); r.y = fmaxf(a.y, b.y);
    r.z = fmaxf(a.z, b.z); r.w = fmaxf(a.w, b.w);
    return r;
}

// CDNA5 async global->LDS copy (tracked by ASYNCcnt, not LOADcnt).
__device__ __forceinline__ void async_load_b128(uint32_t lds_off, const void* gaddr) {
    asm volatile("global_load_async_to_lds_b128 %0, %1, off"
                 :: "v"(lds_off), "v"(gaddr)
                 : "memory");
}
__device__ __forceinline__ void wait_async_zero() {
    asm volatile("s_wait_asynccnt 0" ::: "memory");
}

// ---------------------------------------------------------------------------
// Kernel 1: per-(b,c) spatial sum & max. One block per channel.
// 768 float4 staged through LDS via async copy engine, 16 float4 tail direct.
// ---------------------------------------------------------------------------
__global__ __launch_bounds__(256) void reduce_kernel(const float* __restrict__ x,
                                                     float* __restrict__ avg_out,
                                                     float* __restrict__ max_out) {
    __shared__ v4f   sbuf[NF4_MAIN];   // 12 KB
    __shared__ float wsum[8], wmax[8];

    const int    t    = threadIdx.x;
    const size_t chan = blockIdx.x;                    // b*C + c
    const float* base = x + chan * (size_t)HW;

    const uint32_t lds_base = (uint32_t)(uintptr_t)(&sbuf[0]);
    async_load_b128(lds_base + (uint32_t)(t        ) * 16u, (const void*)(base + (t        ) * 4));
    async_load_b128(lds_base + (uint32_t)(t + 256  ) * 16u, (const void*)(base + (t + 256  ) * 4));
    async_load_b128(lds_base + (uint32_t)(t + 512  ) * 16u, (const void*)(base + (t + 512  ) * 4));

    v4f acc  = {0.f, 0.f, 0.f, 0.f};
    v4f vmax = {-INFINITY, -INFINITY, -INFINITY, -INFINITY};
    if (t < NF4_TAIL) {
        v4f tv = __builtin_nontemporal_load((const v4f*)(base + (NF4_MAIN + t) * 4));
        acc  = tv;
        vmax = tv;
    }

    wait_async_zero();      // this wave's async transfers landed in LDS
    __syncthreads();        // all waves' transfers visible workgroup-wide

    v4f a0 = sbuf[t];
    v4f a1 = sbuf[t + 256];
    v4f a2 = sbuf[t + 512];
    acc  = acc + a0 + a1 + a2;
    vmax = vmax4(vmax4(vmax, a0), vmax4(a1, a2));

    float s = acc.x + acc.y + acc.z + acc.w;
    float m = fmaxf(fmaxf(vmax.x, vmax.y), fmaxf(vmax.z, vmax.w));

    // wave32 shuffle reduction
    #pragma unroll
    for (int off = 16; off > 0; off >>= 1) {
        s += __shfl_xor(s, off, 32);
        m  = fmaxf(m, __shfl_xor(m, off, 32));
    }
    const int wid = t >> 5;
    if ((t & 31) == 0) { wsum[wid] = s; wmax[wid] = m; }
    __syncthreads();

    if (t == 0) {
        float S = 0.f, M = -INFINITY;
        #pragma unroll
        for (int i = 0; i < 8; ++i) { S += wsum[i]; M = fmaxf(M, wmax[i]); }
        avg_out[chan] = S * (1.0f / (float)HW);
        max_out[chan] = M;
    }
}

// ---------------------------------------------------------------------------
// Kernel 2: conv1d(channel axis) + sigmoid + exact top-K with lax.top_k
// tie-break (value desc, index asc), emitted in ascending channel order.
// One block per batch sample, 512 threads (one per channel).
// ---------------------------------------------------------------------------
__global__ __launch_bounds__(512) void score_topk_kernel(const float* __restrict__ avg,
                                                         const float* __restrict__ mx,
                                                         const float* __restrict__ w,
                                                         int*   __restrict__ sidx,
                                                         float* __restrict__ svals) {
    __shared__ float sp[C_DIM + 2];   // zero-padded avg+max descriptor
    __shared__ float score[C_DIM];
    __shared__ int   sel[C_DIM];

    const int c = threadIdx.x;
    const int b = blockIdx.x;

    sp[c + 1] = avg[b * C_DIM + c] + mx[b * C_DIM + c];  // conv(a)+conv(m)=conv(a+m)
    if (c == 0) { sp[0] = 0.f; sp[C_DIM + 1] = 0.f; }
    const float w0 = w[0], w1 = w[1], w2 = w[2];
    __syncthreads();

    // lax.conv is cross-correlation: out[c] = sum_k s[c-1+k] * w[k]
    const float z  = w0 * sp[c] + w1 * sp[c + 1] + w2 * sp[c + 2];
    const float sc = 1.0f / (1.0f + expf(-z));
    score[c] = sc;
    __syncthreads();

    int rank = 0;
    for (int j = 0; j < C_DIM; ++j) {
        const float v = score[j];
        rank += (v > sc) || (v == sc && j < c);
    }
    const int is_sel = (rank < K_SEL) ? 1 : 0;
    sel[c] = is_sel;
    __syncthreads();

    if (is_sel) {
        int pos = 0;
        for (int j = 0; j < c; ++j) pos += sel[j];   // ascending-index compaction
        sidx [b * K_SEL + pos] = c;
        svals[b * K_SEL + pos] = sc;
    }
}

// ---------------------------------------------------------------------------
// Kernel 3: gather selected channels and scale. One block per (b,k).
// Non-temporal b128 traffic: no reuse, keep the 192MB L2 clean.
// ---------------------------------------------------------------------------
__global__ __launch_bounds__(256) void gather_kernel(const float* __restrict__ x,
                                                     const int*   __restrict__ sidx,
                                                     const float* __restrict__ svals,
                                                     float* __restrict__ out) {
    const int p = blockIdx.x;            // b*K + k
    const int t = threadIdx.x;
    const int b = p >> 8;
    const int c = sidx[p];
    const float v = svals[p];

    const float* src = x   + ((size_t)(b * C_DIM + c)) * HW;
    float*       dst = out + (size_t)p * HW;

    #pragma unroll
    for (int i = 0; i < 3; ++i) {
        const int idx = t + i * 256;
        v4f d = __builtin_nontemporal_load((const v4f*)(src + idx * 4));
        d *= v;
        __builtin_nontemporal_store(d, (v4f*)(dst + idx * 4));
    }
    if (t < NF4_TAIL) {
        const int idx = t + NF4_MAIN;
        v4f d = __builtin_nontemporal_load((const v4f*)(src + idx * 4));
        d *= v;
        __builtin_nontemporal_store(d, (v4f*)(dst + idx * 4));
    }
}

extern "C" void kernel_launch(void* const* d_in, const int* in_sizes, int n_in,
                              void* d_out, int out_size, void* d_ws, size_t ws_size,
                              hipStream_t stream) {
    const float* x = (const float*)d_in[0];   // [32,512,56,56]
    const float* w = (const float*)d_in[1];   // [1,1,3]
    float* out = (float*)d_out;               // [32,256,56,56]

    float* avg   = (float*)d_ws;              // B*C
    float* mx    = avg + B_DIM * C_DIM;       // B*C
    int*   sidx  = (int*)(mx + B_DIM * C_DIM);// B*K
    float* svals = (float*)(sidx + B_DIM * K_SEL); // B*K

    reduce_kernel    <<<B_DIM * C_DIM, 256, 0, stream>>>(x, avg, mx);
    score_topk_kernel<<<B_DIM,        C_DIM, 0, stream>>>(avg, mx, w, sidx, svals);
    gather_kernel    <<<B_DIM * K_SEL, 256, 0, stream>>>(x, sidx, svals, out);
}